// SpectralAnalyzer_43413529428414
// MI455X (gfx1250) — compile-verified
//
#include <hip/hip_runtime.h>

typedef float v2f __attribute__((ext_vector_type(2)));
typedef float v8f __attribute__((ext_vector_type(8)));

#define PI2 6.283185307179586f

static __device__ inline v8f wmma4(v2f a, v2f b, v8f c) {
  // D(16x16) = A(16x4) * B(4x16) + C, full f32
  return __builtin_amdgcn_wmma_f32_16x16x4_f32(false, a, false, b, (short)0, c,
                                               false, false);
}

// One workgroup (512 threads = 16 wave32) per 4096-point window.
// Radix-16^3 DIF FFT: each radix-16 stage is W16(16x16) x X(16x256),
// tiled into 16 column-tiles of 16 -> one 16x16 complex matmul per wave
// per level = 16 v_wmma_f32_16x16x4_f32 per tile.
// Input reals staged global->LDS with GLOBAL_LOAD_ASYNC_TO_LDS_B128.
__global__ __launch_bounds__(512) void fft_pow_kernel(const float* __restrict__ x,
                                                      float* __restrict__ out) {
  __shared__ __align__(16) float s_re[4096];
  __shared__ __align__(16) float s_im[4096];
  __shared__ __align__(16) float s_pwr[4096];

  const int tid = threadIdx.x;
  const int w   = blockIdx.x;
  const float* xin = x + (size_t)w * 4096u;

  // ---- async stage: 16KB of reals straight into s_re (ASYNCcnt path) ----
  {
    const uint32_t lds_base = (uint32_t)(uintptr_t)(&s_re[0]);
    const uint64_t gbase    = (uint64_t)(uintptr_t)xin;
#pragma unroll
    for (int it = 0; it < 2; ++it) {
      const int i = tid + it * 512;           // 1024 x 16B = 16KB
      uint32_t loff = lds_base + (uint32_t)i * 16u;
      uint64_t ga   = gbase + (uint64_t)i * 16u;
      asm volatile("global_load_async_to_lds_b128 %0, %1, off"
                   :: "v"(loff), "v"(ga)
                   : "memory");
    }
    // zero imaginary part while the async copy is in flight
    float4* si4 = reinterpret_cast<float4*>(s_im);
    for (int i = tid; i < 1024; i += 512) si4[i] = make_float4(0.f, 0.f, 0.f, 0.f);
  }

  const int lane = tid & 31;
  const int t    = tid >> 5;   // wave id == tile id (wave32)
  const int lo   = lane & 15;
  const int half = lane >> 4;

  // ---- A-matrix = W16[m][k] = exp(-2*pi*i*m*k/16), 4 chunks of 16x4 ----
  // A layout: lane lo = row M; VGPR0 holds K = 4*k0 + 2*half, VGPR1 = +1.
  v2f aRe[4], aIm[4];
#pragma unroll
  for (int k0 = 0; k0 < 4; ++k0) {
    int r0 = 4 * k0 + 2 * half;
    float a0 = (-PI2 / 16.f) * (float)((lo * r0) & 15);
    float a1 = (-PI2 / 16.f) * (float)((lo * (r0 + 1)) & 15);
    float s0, c0, s1, c1;
    __sincosf(a0, &s0, &c0);
    __sincosf(a1, &s1, &c1);
    aRe[k0].x = c0; aRe[k0].y = c1;
    aIm[k0].x = s0; aIm[k0].y = s1;
  }

  asm volatile("s_wait_asynccnt 0x0" ::: "memory");
  __syncthreads();

  // ---- three radix-16 levels ----
#pragma unroll
  for (int lvl = 0; lvl < 3; ++lvl) {
    // element index p(row, col) = base + row*Sr + col*Sc
    const int Sr   = (lvl == 0) ? 256 : (lvl == 1) ? 16 : 1;
    const int Sc   = (lvl == 2) ? 16 : 1;
    const int base = (lvl == 0) ? t * 16 : t * 256;

    v8f accRp = {}, accRn = {}, accI = {};
#pragma unroll
    for (int k0 = 0; k0 < 4; ++k0) {
      int r0 = 4 * k0 + 2 * half;  // B rows for this chunk (lane half mapping)
      int p0 = base + r0 * Sr + lo * Sc;
      int p1 = p0 + Sr;
      v2f bRe, bIm;
      bRe.x = s_re[p0]; bRe.y = s_re[p1];
      bIm.x = s_im[p0]; bIm.y = s_im[p1];
      // (Wr + iWi)(Xr + iXi): Yr = WrXr - WiXi ; Yi = WrXi + WiXr
      accRp = wmma4(aRe[k0], bRe, accRp);
      accRn = wmma4(aIm[k0], bIm, accRn);
      accI  = wmma4(aRe[k0], bIm, accI);
      accI  = wmma4(aIm[k0], bRe, accI);
    }
    v8f yrv = accRp - accRn;

    if (lvl < 2) {
      // twiddle then store back in place (digit replaces its slot)
      const float sc  = (lvl == 0) ? (-PI2 / 4096.f) : (-PI2 / 256.f);
      const int   cix = (lvl == 0) ? (t * 16 + lo) : lo;
      const int   msk = (lvl == 0) ? 4095 : 255;
#pragma unroll
      for (int j = 0; j < 8; ++j) {
        int   M  = half * 8 + j;       // output digit (D row)
        float yr = yrv[j], yi = accI[j];
        float s, c;
        __sincosf(sc * (float)((M * cix) & msk), &s, &c);
        int q = base + M * Sr + lo * Sc;
        s_re[q] = yr * c - yi * s;
        s_im[q] = yr * s + yi * c;
      }
    } else {
      // final level: |.|^2 scattered at digit-reversed index
      // p = t*256 + c*16 + d  ->  k = t + 16*c + 256*d  (c = lo, d = M)
#pragma unroll
      for (int j = 0; j < 8; ++j) {
        int   M  = half * 8 + j;
        float yr = yrv[j], yi = accI[j];
        s_pwr[t + 16 * lo + 256 * M] = yr * yr + yi * yi;
      }
    }
    __syncthreads();
  }

  // ---- coalesced write-out ----
  {
    float4*       o4 = reinterpret_cast<float4*>(out + (size_t)w * 4096u);
    const float4* p4 = reinterpret_cast<const float4*>(s_pwr);
    for (int i = tid; i < 1024; i += 512) o4[i] = p4[i];
  }
}

extern "C" void kernel_launch(void* const* d_in, const int* in_sizes, int n_in,
                              void* d_out, int out_size, void* d_ws, size_t ws_size,
                              hipStream_t stream) {
  const float* x   = (const float*)d_in[0];
  float*       out = (float*)d_out;
  const int    nwin = in_sizes[0] / 4096;   // B = 8192
  fft_pow_kernel<<<dim3(nwin), dim3(512), 0, stream>>>(x, out);
  (void)n_in; (void)out_size; (void)d_ws; (void)ws_size;
}